// _SE_49684181680676
// MI455X (gfx1250) — compile-verified
//
#include <hip/hip_runtime.h>
#include <math.h>

typedef float v2f __attribute__((ext_vector_type(2)));
typedef float v4f __attribute__((ext_vector_type(4)));
typedef float v8f __attribute__((ext_vector_type(8)));

#define SE_CH    64
#define SE_M     16
#define SE_B     32
#define SE_HW    16384   // 128*128
#define SE_HW4   4096    // float4 per plane

// ---------------------------------------------------------------------------
// Kernel 1: per-(b,c) plane sum. One 256-thread block per plane, float4 loads.
// Regular-temporal loads on purpose: x (128 MiB) fits in the 192 MB L2, so
// this pass primes L2 for the scale pass.
// ---------------------------------------------------------------------------
__global__ void se_pool_sum(const float* __restrict__ x, float* __restrict__ sums) {
    __shared__ float red[256];
    const int plane = blockIdx.x;            // b*64 + c, 0..2047
    const int tid   = threadIdx.x;
    const v4f* p = reinterpret_cast<const v4f*>(x) + (size_t)plane * SE_HW4;
    float acc = 0.f;
#pragma unroll
    for (int j = 0; j < 16; ++j) {           // 16 * 256 = 4096 float4 = 16384 floats
        v4f v = p[tid + j * 256];
        acc += (v.x + v.y) + (v.z + v.w);
    }
    red[tid] = acc;
    __syncthreads();
#pragma unroll
    for (int s = 128; s > 0; s >>= 1) {
        if (tid < s) red[tid] += red[tid + s];
        __syncthreads();
    }
    if (tid == 0) sums[plane] = red[0];
}

// ---------------------------------------------------------------------------
// Kernel 2: gate computation with V_WMMA_F32_16X16X4_F32 (CDNA5 fp32 WMMA).
// One full wave32 (EXEC all ones, required by WMMA).
//   H[32x16] = silu( (sums/HW)[32x64] @ w1[16x64]^T + b1 )
//   G[32x64] = sigmoid( H[32x16] @ w2[64x16]^T + b2 )
// Fragment layouts per ISA 7.12.2:
//   A 16x4 f32 : lane l (hf=l>>4, id=l&15): a[0]=A[id][k0+2*hf], a[1]=A[id][k0+2*hf+1]
//   B 4x16 f32 : b[0]=B[k0+2*hf][id], b[1]=B[k0+2*hf+1][id]
//   C/D 16x16  : acc[v] at lane l -> (M = v + 8*hf, N = id)
// ---------------------------------------------------------------------------
__global__ void se_gate(const float* __restrict__ sums,
                        const float* __restrict__ w1, const float* __restrict__ b1,
                        const float* __restrict__ w2, const float* __restrict__ b2,
                        float* __restrict__ g) {
    __shared__ float ldsH[SE_B * SE_M];      // H staged D-layout -> A-layout
    const int lane = threadIdx.x;            // 0..31
    const int id   = lane & 15;
    const int hf   = lane >> 4;
    const float inv = 1.0f / (float)SE_HW;

    // ---- matmul 1: H = silu(S @ W1^T + b1), two 16-row batch tiles ----
#pragma unroll
    for (int t = 0; t < 2; ++t) {
        v8f acc = {};
        const int row = t * 16 + id;         // batch row for A-frag
#pragma unroll
        for (int k0 = 0; k0 < SE_CH; k0 += 4) {
            const int k = k0 + 2 * hf;
            v2f a, b;
            a.x = sums[row * SE_CH + k]     * inv;     // A = S (mean)
            a.y = sums[row * SE_CH + k + 1] * inv;
            b.x = w1[id * SE_CH + k];                  // B[k][n] = w1[n][k]
            b.y = w1[id * SE_CH + k + 1];
            acc = __builtin_amdgcn_wmma_f32_16x16x4_f32(
                false, a, false, b, (short)0, acc, false, false);
        }
#pragma unroll
        for (int v = 0; v < 8; ++v) {
            const int br  = t * 16 + v + 8 * hf;       // batch row of D element
            const float pre = acc[v] + b1[id];
            const float sg  = 1.0f / (1.0f + __expf(-pre));
            ldsH[br * SE_M + id] = pre * sg;           // silu
        }
    }
    __syncthreads();

    // ---- matmul 2: G = sigmoid(H @ W2^T + b2), 2 batch tiles x 4 channel tiles ----
#pragma unroll
    for (int t = 0; t < 2; ++t) {
#pragma unroll
        for (int n = 0; n < 4; ++n) {
            v8f acc = {};
#pragma unroll
            for (int k0 = 0; k0 < SE_M; k0 += 4) {
                const int k = k0 + 2 * hf;
                v2f a, b;
                a.x = ldsH[(t * 16 + id) * SE_M + k];          // A = H
                a.y = ldsH[(t * 16 + id) * SE_M + k + 1];
                b.x = w2[(n * 16 + id) * SE_M + k];            // B[k][c] = w2[c][k]
                b.y = w2[(n * 16 + id) * SE_M + k + 1];
                acc = __builtin_amdgcn_wmma_f32_16x16x4_f32(
                    false, a, false, b, (short)0, acc, false, false);
            }
#pragma unroll
            for (int v = 0; v < 8; ++v) {
                const int br = t * 16 + v + 8 * hf;
                const int c  = n * 16 + id;
                const float pre = acc[v] + b2[c];
                g[br * SE_CH + c] = 1.0f / (1.0f + __expf(-pre));
            }
        }
    }
}

// ---------------------------------------------------------------------------
// Kernel 3: out = x * g[plane], float4 vectorized. x read should hit L2
// (primed by pass 1); output uses non-temporal stores so the 128 MiB write
// stream does not evict x from L2.
// ---------------------------------------------------------------------------
__global__ void se_scale(const float* __restrict__ x, const float* __restrict__ g,
                         float* __restrict__ out) {
    const size_t i = (size_t)blockIdx.x * blockDim.x + threadIdx.x;  // float4 index
    const int plane = (int)(i >> 12);                                // /4096
    const float s = g[plane];
    v4f v = reinterpret_cast<const v4f*>(x)[i];
    v4f r;
    r.x = v.x * s; r.y = v.y * s; r.z = v.z * s; r.w = v.w * s;
    __builtin_nontemporal_store(r, reinterpret_cast<v4f*>(out) + i);
}

// ---------------------------------------------------------------------------
extern "C" void kernel_launch(void* const* d_in, const int* in_sizes, int n_in,
                              void* d_out, int out_size, void* d_ws, size_t ws_size,
                              hipStream_t stream) {
    (void)in_sizes; (void)n_in; (void)out_size; (void)ws_size;
    const float* x  = (const float*)d_in[0];
    const float* w1 = (const float*)d_in[1];
    const float* b1 = (const float*)d_in[2];
    const float* w2 = (const float*)d_in[3];
    const float* b2 = (const float*)d_in[4];
    float* out = (float*)d_out;

    float* sums = (float*)d_ws;                      // 2048 floats
    float* g    = (float*)d_ws + 2048;               // 2048 floats

    // pass 1: pooled sums (one block per plane)
    se_pool_sum<<<SE_B * SE_CH, 256, 0, stream>>>(x, sums);
    // pass 2: gates via fp32 WMMA (single wave)
    se_gate<<<1, 32, 0, stream>>>(sums, w1, b1, w2, b2, g);
    // pass 3: scale (8M float4 elements)
    const int total4 = SE_B * SE_CH * SE_HW4;        // 8,388,608
    se_scale<<<total4 / 256, 256, 0, stream>>>(x, g, out);
}